// Kuramoto2D_8512625180941
// MI455X (gfx1250) — compile-verified
//
#include <hip/hip_runtime.h>

// ---- problem constants -----------------------------------------------------
constexpr int   Hh     = 224;
constexpr int   Ww     = 224;
constexpr int   NSTEPS = 10;
constexpr float DTc    = 0.15f;
constexpr float DEBIAS = 1.000244140625f;       // 1 + 2^-12: makes RTZ f16 pack ~RTN

constexpr int PITCH     = Ww + 2;               // 226 (zero border => free zero-padding)
constexpr int SC_WORDS  = (Hh + 2) * (Ww + 2);  // 51076 packed f16x2 cells
constexpr int STAGE_W   = 51200;                // word offset of staging region (pad/align)
constexpr int CHUNK_ROWS  = 112;
constexpr int CHUNK_ELEMS = CHUNK_ROWS * Ww;    // 25088 floats per async chunk
constexpr int SMEM_BYTES  = STAGE_W * 4 + CHUNK_ELEMS * 4;  // 204800 + 100352 = 305152 <= 320K

typedef _Float16 h2 __attribute__((ext_vector_type(2)));

static __device__ __forceinline__ unsigned packSC(float s, float c) {
  auto h = __builtin_amdgcn_cvt_pkrtz(s, c);    // v_cvt_pk_rz_f16_f32 -> ONE live VGPR
  return __builtin_bit_cast(unsigned, h);
}
static __device__ __forceinline__ void unpackSC(unsigned u, float& s, float& c) {
  h2 h = __builtin_bit_cast(h2, u);
  s = (float)h.x;
  c = (float)h.y;
}
static __device__ __forceinline__ unsigned addh2(unsigned a, unsigned b) {
  h2 x = __builtin_bit_cast(h2, a);
  h2 y = __builtin_bit_cast(h2, b);
  h2 r = x + y;                                  // v_pk_add_f16
  return __builtin_bit_cast(unsigned, r);
}

__global__ __launch_bounds__(1024, 1)
void kuramoto2d_kernel(const float* __restrict__ x,
                       const float* __restrict__ om,
                       const float* __restrict__ Kp,
                       float* __restrict__ out)
{
  extern __shared__ unsigned smem[];
  unsigned*  sc      = smem;                                   // packed (sin,cos) field, padded
  float*     xstage  = reinterpret_cast<float*>(smem + STAGE_W);
  _Float16*  domega  = reinterpret_cast<_Float16*>(smem + STAGE_W); // reused after init

  const int tid  = threadIdx.x;
  const int lane = tid & 31;
  const int wave = tid >> 5;
  const int r0   = wave * 7;            // 32 waves x 7 rows = 224 rows
  const int b    = blockIdx.x;

  // ---- zero the whole packed field (border stays 0 => zero-padding) --------
  for (int t = tid; t < SC_WORDS; t += 1024) sc[t] = 0u;

  // ---- init: async-load x in two 112-row chunks, convert to (sin,cos) -----
  for (int ch = 0; ch < 2; ++ch) {
    const float* src = x + (size_t)b * (Hh * Ww) + (size_t)ch * CHUNK_ELEMS;
#pragma unroll 1
    for (int k = 0; k < 7; ++k) {
      int v4 = k * 1024 + tid;                   // 16B vector index
      if (v4 < CHUNK_ELEMS / 4) {
        unsigned ldsb = (unsigned)(STAGE_W * 4) + (unsigned)v4 * 16u;
        unsigned goff = (unsigned)v4 * 16u;
        asm volatile("global_load_async_to_lds_b128 %0, %1, %2"
                     :: "v"(ldsb), "v"(goff), "s"(src) : "memory");
      }
    }
    asm volatile("s_wait_asynccnt 0" ::: "memory");
    __syncthreads();                             // all waves' async copies + zeroing done
#pragma unroll 1
    for (int k = 0; k < 25; ++k) {
      int e = k * 1024 + tid;
      if (e < CHUNK_ELEMS) {
        float xv = xstage[e];
        float th = fmaf(6.28318530717958648f, xv, -3.14159265358979324f); // pi*(2x-1)
        float s0 = __sinf(th) * DEBIAS;
        float c0 = __cosf(th) * DEBIAS;
        int r = ch * CHUNK_ROWS + e / Ww;
        int c = e - (e / Ww) * Ww;
        sc[(r + 1) * PITCH + (c + 1)] = packSC(s0, c0);
      }
    }
    __syncthreads();
  }

  // ---- cache DT*omega as f16 in the (now free) staging region --------------
  const float Kv  = Kp[0];
  const float dtk = DTc * Kv;
#pragma unroll 1
  for (int k = 0; k < 49; ++k) {
    int e = k * 1024 + tid;                      // 49*1024 == 224*224 exactly
    domega[e] = (_Float16)(DTc * om[e]);
  }
  __syncthreads();

  // ---- 10 Jacobi steps, fully LDS-resident ---------------------------------
  unsigned regs[49];                             // live only phase A -> phase B
#pragma unroll 1
  for (int step = 0; step < NSTEPS; ++step) {
    // phase A: compute new packed (sin,cos) into registers
#pragma unroll
    for (int i = 0; i < 7; ++i) {
      const int r    = r0 + i;
      const int base = (r + 1) * PITCH;
#pragma unroll
      for (int j = 0; j < 7; ++j) {
        const int c   = lane + 32 * j;
        const int idx = base + c + 1;
        unsigned pm = sc[idx];
        unsigned nb = addh2(addh2(sc[idx - PITCH], sc[idx + PITCH]),
                            addh2(sc[idx - 1],     sc[idx + 1]));
        float s, cc, sn, cn;
        unpackSC(pm, s, cc);
        unpackSC(nb, sn, cn);
        float coup  = fmaf(cc, sn, -(s * cn));          // cos*sin_n - sin*cos_n
        float dtw   = (float)domega[r * Ww + c];
        float delta = fmaf(dtk, coup, dtw);             // DT*(omega + K*coupling)
        float sd = __sinf(delta) * DEBIAS;              // scaled rotation => outputs
        float cd = __cosf(delta) * DEBIAS;              //  pre-biased for RTZ pack
        float ns = fmaf(s,  cd,  cc * sd);              // ~sin(theta+delta)*DEBIAS
        float nc = fmaf(cc, cd, -(s * sd));             // ~cos(theta+delta)*DEBIAS
        regs[i * 7 + j] = packSC(ns, nc);
      }
    }
    __syncthreads();
    // phase B: publish new values for neighbor reads
#pragma unroll
    for (int i = 0; i < 7; ++i) {
      const int base = (r0 + i + 1) * PITCH + 1 + lane;
#pragma unroll
      for (int j = 0; j < 7; ++j) sc[base + 32 * j] = regs[i * 7 + j];
    }
    __syncthreads();
  }

  // ---- emit features [cos ; sin] straight from last-step registers ---------
  const size_t ob = (size_t)b * (size_t)(2 * Hh * Ww);
#pragma unroll
  for (int i = 0; i < 7; ++i) {
    const int r = r0 + i;
#pragma unroll
    for (int j = 0; j < 7; ++j) {
      const int c   = lane + 32 * j;
      const int off = r * Ww + c;
      float s, cc;
      unpackSC(regs[i * 7 + j], s, cc);
      out[ob + off]           = cc;   // channel 0: cos
      out[ob + Hh * Ww + off] = s;    // channel 1: sin
    }
  }
}

extern "C" void kernel_launch(void* const* d_in, const int* in_sizes, int n_in,
                              void* d_out, int out_size, void* d_ws, size_t ws_size,
                              hipStream_t stream) {
  (void)n_in; (void)out_size; (void)d_ws; (void)ws_size;
  const float* x  = (const float*)d_in[0];
  const float* om = (const float*)d_in[1];
  const float* K  = (const float*)d_in[2];
  float* out = (float*)d_out;

  const int B = in_sizes[0] / (Hh * Ww);   // 256

  (void)hipFuncSetAttribute((const void*)kuramoto2d_kernel,
                            hipFuncAttributeMaxDynamicSharedMemorySize, SMEM_BYTES);
  kuramoto2d_kernel<<<B, 1024, SMEM_BYTES, stream>>>(x, om, K, out);
}